// PointNetFeatures_89661737271401
// MI455X (gfx1250) — compile-verified
//
#include <hip/hip_runtime.h>
#include <math.h>

#define NPTS  4096
#define BATCH 4
#define HEADS 4
#define CCH   16
#define KNN   40
#define INV_BN 0.9999950000374998f   // 1/sqrt(1+1e-5)

typedef __attribute__((ext_vector_type(2))) float v2f;
typedef __attribute__((ext_vector_type(8))) float v8f;

__device__ __forceinline__ float leaky(float v) { return v >= 0.f ? v : 0.01f * v; }

// ---------------------------------------------------------------------------
// Kernel 1: kNN top-K per point. One workgroup per (b,n). Distance row lives
// in LDS (16KB). K iterations of tie-broken argmax (lowest index on equality,
// matching lax.top_k). Wave-level shfl_xor reduction (wave32) + one 8-entry
// cross-wave merge -> only 2 barriers per iteration.
// ---------------------------------------------------------------------------
__global__ void knn_topk_kernel(const float* __restrict__ x, int* __restrict__ idx_out) {
  __shared__ float pd[NPTS];
  __shared__ float rv[8];
  __shared__ int   ri[8];
  const int bn = blockIdx.x;             // b*NPTS + n
  const int b  = bn >> 12;
  const int n  = bn & (NPTS - 1);
  const float* xb = x + (size_t)b * 3 * NPTS;
  const float xn0 = xb[n], xn1 = xb[NPTS + n], xn2 = xb[2 * NPTS + n];
  const float sqn = xn0 * xn0 + xn1 * xn1 + xn2 * xn2;
  for (int m = threadIdx.x; m < NPTS; m += 256) {
    float m0 = xb[m], m1 = xb[NPTS + m], m2 = xb[2 * NPTS + m];
    pd[m] = 2.f * (xn0 * m0 + xn1 * m1 + xn2 * m2) - sqn - (m0 * m0 + m1 * m1 + m2 * m2);
  }
  __syncthreads();
  int* out = idx_out + (size_t)bn * KNN;
  const int wv = threadIdx.x >> 5;
  for (int k = 0; k < KNN; ++k) {
    float bv = -__builtin_inff(); int bi = NPTS;
    for (int m = threadIdx.x; m < NPTS; m += 256) {
      float v = pd[m];
      if (v > bv) { bv = v; bi = m; }    // strict > keeps lowest index in chunk
    }
    // wave32 reduction with lowest-index tie-break
#pragma unroll
    for (int off = 16; off > 0; off >>= 1) {
      float ov = __shfl_xor(bv, off, 32);
      int   oi = __shfl_xor(bi, off, 32);
      if (ov > bv || (ov == bv && oi < bi)) { bv = ov; bi = oi; }
    }
    if ((threadIdx.x & 31) == 0) { rv[wv] = bv; ri[wv] = bi; }
    __syncthreads();
    if (threadIdx.x == 0) {
      float fv = rv[0]; int fi = ri[0];
#pragma unroll
      for (int w = 1; w < 8; ++w) {
        float ov = rv[w]; int oi = ri[w];
        if (ov > fv || (ov == fv && oi < fi)) { fv = ov; fi = oi; }
      }
      out[k] = fi;
      pd[fi] = -__builtin_inff();
    }
    __syncthreads();
  }
}

// ---------------------------------------------------------------------------
// Kernel 2: fused GAP heads. One thread per (b,h,n). Online softmax over K so
// xl (B,H,N,K,16) is never materialized. Emits xg_cat and xl_max in (B,64,N).
// ---------------------------------------------------------------------------
__global__ void gap_kernel(
    const float* __restrict__ x, const int* __restrict__ knn_idx,
    const float* __restrict__ W1, const float* __restrict__ g1, const float* __restrict__ be1,
    const float* __restrict__ W2, const float* __restrict__ g2, const float* __restrict__ be2,
    const float* __restrict__ W3, const float* __restrict__ g3, const float* __restrict__ be3,
    const float* __restrict__ W4, const float* __restrict__ g4, const float* __restrict__ be4,
    float* __restrict__ xg_cat, float* __restrict__ xl_max) {
  const int g  = blockIdx.x * blockDim.x + threadIdx.x;  // (b*H + h)*NPTS + n
  const int n  = g & (NPTS - 1);
  const int bh = g >> 12;
  const int h  = bh & (HEADS - 1);
  const int b  = bh >> 2;
  const float* xb = x + (size_t)b * 3 * NPTS;
  const float p0 = xb[n], p1 = xb[NPTS + n], p2 = xb[2 * NPTS + n];

  // ---- xg scalar: leaky(bn(W2 . leaky(bn(W1 @ p)))) ----
  const float* W1h = W1 + h * CCH * 3;
  const float* g1h = g1 + h * CCH;
  const float* be1h = be1 + h * CCH;
  const float* W2h = W2 + h * CCH;
  float u = 0.f;
#pragma unroll
  for (int c = 0; c < CCH; ++c) {
    float v = W1h[c * 3] * p0 + W1h[c * 3 + 1] * p1 + W1h[c * 3 + 2] * p2;
    v = leaky(v * (g1h[c] * INV_BN) + be1h[c]);
    u += W2h[c] * v;
  }
  const float xg = leaky(u * (g2[h] * INV_BN) + be2[h]);

  // ---- neighbor loop: xl rows, xk, online softmax-weighted sum, max ----
  const float* W3h = W3 + h * CCH * 3;
  const float* g3h = g3 + h * CCH;
  const float* be3h = be3 + h * CCH;
  const float* W4h = W4 + h * CCH;
  const float g4h = g4[h] * INV_BN, be4h = be4[h];

  float F[CCH], mx[CCH];
#pragma unroll
  for (int c = 0; c < CCH; ++c) { F[c] = 0.f; mx[c] = -__builtin_inff(); }
  float S = 0.f, M = -__builtin_inff();
  const int* nb = knn_idx + ((size_t)(b * NPTS + n)) * KNN;
  for (int k = 0; k < KNN; ++k) {
    const int mi = nb[k];
    const float c0 = xb[mi], c1 = xb[NPTS + mi], c2 = xb[2 * NPTS + mi];
    float xl[CCH]; float u4 = 0.f;
#pragma unroll
    for (int c = 0; c < CCH; ++c) {
      float v = W3h[c * 3] * c0 + W3h[c * 3 + 1] * c1 + W3h[c * 3 + 2] * c2;
      v = leaky(v * (g3h[c] * INV_BN) + be3h[c]);
      xl[c] = v; mx[c] = fmaxf(mx[c], v); u4 += W4h[c] * v;
    }
    const float s  = leaky(xg + leaky(u4 * g4h + be4h));
    const float nM = fmaxf(M, s);
    const float sc = __expf(M - nM);   // 0 on first iteration (M = -inf)
    const float w  = __expf(s - nM);
#pragma unroll
    for (int c = 0; c < CCH; ++c) F[c] = F[c] * sc + w * xl[c];
    S = S * sc + w;
    M = nM;
  }
  const float invS = 1.f / S;
  float* og = xg_cat + ((size_t)b * 64 + h * CCH) * NPTS + n;
  float* om = xl_max + ((size_t)b * 64 + h * CCH) * NPTS + n;
#pragma unroll
  for (int c = 0; c < CCH; ++c) { og[c * NPTS] = F[c] * invS; om[c * NPTS] = mx[c]; }
}

// ---------------------------------------------------------------------------
// Kernel 3: 1x1 conv (64 out channels) via V_WMMA_F32_16X16X4_F32.
// Templated on channel counts and epilogue so the K-loop fully unrolls with
// compile-time boundary handling. Each wave owns TWO adjacent 16-column tiles
// (16x32 output, 8 v8f accumulators) and all four 16-row tiles, so each
// A (weight) fragment is reused across 2 WMMAs and each B fragment across 4.
// Input channels = [in0 (C0 ch) | in1 (C1 ch)], both laid out (B, Cx, NPTS).
// ---------------------------------------------------------------------------
template <int C0, int C1, bool BN, bool RELU>
__global__ void conv_wmma_kernel(
    const float* __restrict__ in0, const float* __restrict__ in1,
    const float* __restrict__ W, const float* __restrict__ bias,
    const float* __restrict__ gamma, const float* __restrict__ beta,
    float* __restrict__ out) {
  constexpr int Ktot = C0 + C1;
  const int lane = threadIdx.x & 31;
  const int wid  = blockIdx.x * (blockDim.x >> 5) + (threadIdx.x >> 5);
  const int b    = wid >> 7;                 // 128 column tile-pairs per batch
  const int tp   = wid & 127;
  const int ncol = tp * 32 + (lane & 15);    // first tile's column for this lane
  const int kg   = (lane >> 4) << 1;         // 0 for lanes 0-15, 2 for 16-31
  const int mrow = lane & 15;                // A-fragment row within tile

  auto ldin = [&](int k, int col) -> float {
    if constexpr (C1 == 0) {
      return in0[((size_t)b * C0 + k) * NPTS + col];
    } else {
      return (k < C0) ? in0[((size_t)b * C0 + k) * NPTS + col]
                      : in1[((size_t)b * C1 + (k - C0)) * NPTS + col];
    }
  };

  v8f acc[2][4];
#pragma unroll
  for (int t = 0; t < 2; ++t)
#pragma unroll
    for (int ot = 0; ot < 4; ++ot)
      acc[t][ot] = (v8f){0.f, 0.f, 0.f, 0.f, 0.f, 0.f, 0.f, 0.f};

#pragma unroll
  for (int k0 = 0; k0 < Ktot; k0 += 4) {
    const int ka = k0 + kg;
    v2f bf0, bf1;
    if (k0 + 4 <= Ktot) {                    // compile-time per unrolled chunk
      bf0.x = ldin(ka, ncol);          bf0.y = ldin(ka + 1, ncol);
      bf1.x = ldin(ka, ncol + 16);     bf1.y = ldin(ka + 1, ncol + 16);
    } else {                                 // boundary chunk (K=67 only)
      bf0.x = (ka     < Ktot) ? ldin(ka,     ncol)      : 0.f;
      bf0.y = (ka + 1 < Ktot) ? ldin(ka + 1, ncol)      : 0.f;
      bf1.x = (ka     < Ktot) ? ldin(ka,     ncol + 16) : 0.f;
      bf1.y = (ka + 1 < Ktot) ? ldin(ka + 1, ncol + 16) : 0.f;
    }
#pragma unroll
    for (int ot = 0; ot < 4; ++ot) {
      const int row = ot * 16 + mrow;
      v2f af;
      if (k0 + 4 <= Ktot) {
        af.x = W[(size_t)row * Ktot + ka];
        af.y = W[(size_t)row * Ktot + ka + 1];
      } else {
        af.x = (ka     < Ktot) ? W[(size_t)row * Ktot + ka]     : 0.f;
        af.y = (ka + 1 < Ktot) ? W[(size_t)row * Ktot + ka + 1] : 0.f;
      }
      acc[0][ot] = __builtin_amdgcn_wmma_f32_16x16x4_f32(
          false, af, false, bf0, (short)0, acc[0][ot], false, false);
      acc[1][ot] = __builtin_amdgcn_wmma_f32_16x16x4_f32(
          false, af, false, bf1, (short)0, acc[1][ot], false, false);
    }
  }

  // Epilogue: D VGPR r -> row r (lanes 0-15) / r+8 (lanes 16-31), col = lane%16
  const int half8 = (lane >> 4) * 8;
#pragma unroll
  for (int t = 0; t < 2; ++t) {
#pragma unroll
    for (int ot = 0; ot < 4; ++ot) {
#pragma unroll
      for (int r = 0; r < 8; ++r) {
        const int o = ot * 16 + half8 + r;
        float v = acc[t][ot][r] + bias[o];
        if constexpr (BN)   v = v * (gamma[o] * INV_BN) + beta[o];
        if constexpr (RELU) v = fmaxf(v, 0.f);
        out[((size_t)b * 64 + o) * NPTS + ncol + t * 16] = v;
      }
    }
  }
}

// ---------------------------------------------------------------------------
extern "C" void kernel_launch(void* const* d_in, const int* in_sizes, int n_in,
                              void* d_out, int out_size, void* d_ws, size_t ws_size,
                              hipStream_t stream) {
  (void)in_sizes; (void)n_in; (void)out_size; (void)ws_size;
  const float* x    = (const float*)d_in[0];
  const float* W1   = (const float*)d_in[1];
  const float* g1   = (const float*)d_in[2];
  const float* be1  = (const float*)d_in[3];
  const float* W2   = (const float*)d_in[4];
  const float* g2   = (const float*)d_in[5];
  const float* be2  = (const float*)d_in[6];
  const float* W3   = (const float*)d_in[7];
  const float* g3   = (const float*)d_in[8];
  const float* be3  = (const float*)d_in[9];
  const float* W4   = (const float*)d_in[10];
  const float* g4   = (const float*)d_in[11];
  const float* be4  = (const float*)d_in[12];
  const float* Wc1  = (const float*)d_in[13];
  const float* bc1  = (const float*)d_in[14];
  const float* gc1  = (const float*)d_in[15];
  const float* bnb1 = (const float*)d_in[16];
  const float* Wc2  = (const float*)d_in[17];
  const float* bc2  = (const float*)d_in[18];
  const float* gc2  = (const float*)d_in[19];
  const float* bnb2 = (const float*)d_in[20];
  const float* Wc3  = (const float*)d_in[21];
  const float* bc3  = (const float*)d_in[22];

  // Workspace layout (~18.6 MB total)
  char* ws = (char*)d_ws;
  size_t off = 0;
  auto take = [&](size_t bytes) -> void* {
    void* p = ws + off;
    off += bytes;
    off = (off + 255) & ~(size_t)255;
    return p;
  };
  int*   knn = (int*)  take((size_t)BATCH * NPTS * KNN * sizeof(int));
  float* xgc = (float*)take((size_t)BATCH * 64 * NPTS * sizeof(float));
  float* xlm = (float*)take((size_t)BATCH * 64 * NPTS * sizeof(float));
  float* h1  = (float*)take((size_t)BATCH * 64 * NPTS * sizeof(float));
  float* h2  = (float*)take((size_t)BATCH * 64 * NPTS * sizeof(float));

  // 1) kNN graph: one workgroup per point
  knn_topk_kernel<<<BATCH * NPTS, 256, 0, stream>>>(x, knn);

  // 2) fused GAP heads: one thread per (b,h,n)
  gap_kernel<<<(BATCH * HEADS * NPTS) / 256, 256, 0, stream>>>(
      x, knn, W1, g1, be1, W2, g2, be2, W3, g3, be3, W4, g4, be4, xgc, xlm);

  // 3-5) WMMA convs: 512 waves total = 64 blocks x 8 waves (2 tiles/wave)
  conv_wmma_kernel<64, 3, true, true><<<64, 256, 0, stream>>>(
      xgc, x, Wc1, bc1, gc1, bnb1, h1);
  conv_wmma_kernel<64, 0, true, true><<<64, 256, 0, stream>>>(
      h1, nullptr, Wc2, bc2, gc2, bnb2, h2);
  conv_wmma_kernel<64, 64, false, false><<<64, 256, 0, stream>>>(
      h2, xlm, Wc3, bc3, nullptr, nullptr, (float*)d_out);
}